// Model_82583631167661
// MI455X (gfx1250) — compile-verified
//
#include <hip/hip_runtime.h>
#include <math.h>

// Problem constants (from reference setup_inputs)
#define B  4
#define T  288
#define V  512
#define C  32
#define E  3
#define EV 8
#define W  60

typedef __attribute__((ext_vector_type(2))) float v2f;
typedef __attribute__((ext_vector_type(8))) float v8f;

// ---------------------------------------------------------------------------
// Kernel 1: expert weights  ew = softmax(sigmoid(relu(ef@w1+b1)@w2+b2))
// ---------------------------------------------------------------------------
__global__ void ew_kernel(const float* __restrict__ ef,
                          const float* __restrict__ w1, const float* __restrict__ b1,
                          const float* __restrict__ w2, const float* __restrict__ b2,
                          float* __restrict__ ew) {
    __shared__ float h[B][32];
    __shared__ float sg[B][5];
    const int tid = threadIdx.x;             // 32 threads
    if (tid < 32) {
        for (int b = 0; b < B; ++b) {
            float acc = b1[tid];
            #pragma unroll
            for (int k = 0; k < EV; ++k) acc += ef[b * EV + k] * w1[k * 32 + tid];
            h[b][tid] = fmaxf(acc, 0.0f);
        }
    }
    __syncthreads();
    if (tid < 5) {
        for (int b = 0; b < B; ++b) {
            float acc = b2[tid];
            for (int k = 0; k < 32; ++k) acc += h[b][k] * w2[k * 5 + tid];
            sg[b][tid] = 1.0f / (1.0f + expf(-acc));
        }
    }
    __syncthreads();
    if (tid < B) {
        const int b = tid;
        float mx = sg[b][0];
        #pragma unroll
        for (int i = 1; i < 5; ++i) mx = fmaxf(mx, sg[b][i]);
        float e[5], sum = 0.0f;
        #pragma unroll
        for (int i = 0; i < 5; ++i) { e[i] = expf(sg[b][i] - mx); sum += e[i]; }
        #pragma unroll
        for (int i = 0; i < 5; ++i) ew[b * 5 + i] = e[i] / sum;
    }
}

// ---------------------------------------------------------------------------
// Kernel 2: impact_sum[b,t] = sum_e alpha_e * reversed decayed cumsum
// ---------------------------------------------------------------------------
__global__ void impact_kernel(const float* __restrict__ events,
                              float* __restrict__ imp) {
    const int b = threadIdx.x;               // 32 threads, 4 active
    if (b >= B) return;
    __shared__ float acc[B][T];
    for (int t = 0; t < T; ++t) acc[b][t] = 0.0f;
    const float decay[E] = {300.0f, 600.0f, 180.0f};
    const float alpha[E] = {0.3f, 0.5f, 0.2f};
    for (int e = 0; e < E; ++e) {
        const float r = expf(-1.0f / decay[e]);
        float carry = 0.0f;
        for (int t = T - 1; t >= 0; --t) {
            carry = events[(b * T + t) * E + e] + r * carry;
            acc[b][t] += alpha[e] * carry;
        }
    }
    for (int t = 0; t < T; ++t) imp[b * T + t] = acc[b][t];
}

// ---------------------------------------------------------------------------
// Kernel 3: cv[b,t,v] = rolling (Bessel) std/mean over W=60, left-padded 0
//   one thread per (b,v); LDS ring of 60 per thread
// ---------------------------------------------------------------------------
__global__ void cv_kernel(const float* __restrict__ flow,
                          float* __restrict__ cvout) {
    __shared__ float ring[W][256];
    const int tid = threadIdx.x;
    const int g = blockIdx.x * 256 + tid;    // over B*V
    const int v = g % V;
    const int b = g / V;
    const float* fs = flow + (size_t)b * T * V + v;
    float* co = cvout + (size_t)b * T * V + v;
    float s = 0.0f, s2 = 0.0f;
    for (int t = 0; t < T; ++t) {
        const float nv = fs[(size_t)t * V];
        const int slot = t % W;
        const float old = (t >= W) ? ring[slot][tid] : 0.0f;
        ring[slot][tid] = nv;
        s  += nv - old;
        s2 += nv * nv - old * old;
        float res = 0.0f;
        if (t >= W - 1) {
            const float mean = s * (1.0f / W);
            float var = (s2 - (float)W * mean * mean) * (1.0f / (W - 1));
            var = fmaxf(var, 0.0f);
            res = sqrtf(var) / (mean + 1e-6f);
        }
        co[(size_t)t * V] = res;
    }
}

// ---------------------------------------------------------------------------
// Kernel 4: combined[b,c,t,v] = sum_i (ew[b,i]/k_i) * windowsum_i(x[b,c,:,v])
//   window_i(t) = [t - k/2, t + k - k/2 - 1]  (zeros outside, /k always)
//   experts k = {3,6,12,24,48}, p=k/2 = {1,3,6,12,24}
//   incremental scan: add x[t+{2,3,6,12,24}], sub x[t-{1,3,6,12,24}]
//   ring span needed: [t-24, t+24] relative -> 49 slots
// ---------------------------------------------------------------------------
__global__ void comb_kernel(const float* __restrict__ x,
                            const float* __restrict__ ew,
                            float* __restrict__ comb) {
    __shared__ float ring[49][256];
    const int tid = threadIdx.x;
    const int g = blockIdx.x * 256 + tid;    // over B*C*V
    const int v = g % V;
    const int c = (g / V) % C;
    const int b = g / (V * C);
    const float* xs = x   + ((size_t)(b * C + c) * T) * V + v;
    float*       cs = comb + ((size_t)(b * C + c) * T) * V + v;

    const float kf[5] = {3.0f, 6.0f, 12.0f, 24.0f, 48.0f};
    float wgt[5];
    #pragma unroll
    for (int i = 0; i < 5; ++i) wgt[i] = ew[b * 5 + i] / kf[i];

    float s0 = 0, s1 = 0, s2 = 0, s3 = 0, s4 = 0;
    // prologue: indices 0..23 prime the ring and the t=0 window sums
    for (int j = 0; j < 24; ++j) {
        const float val = xs[(size_t)j * V];
        ring[j % 49][tid] = val;
        if (j < 2)  s0 += val;
        if (j < 3)  s1 += val;
        if (j < 6)  s2 += val;
        if (j < 12) s3 += val;
        s4 += val;                            // j < 24 always here
    }
    for (int t = 0; t < T; ++t) {
        cs[(size_t)t * V] = wgt[0]*s0 + wgt[1]*s1 + wgt[2]*s2 + wgt[3]*s3 + wgt[4]*s4;
        // advance windows to t+1
        const int jn = t + 24;
        const float nv = (jn < T) ? xs[(size_t)jn * V] : 0.0f;
        ring[jn % 49][tid] = nv;
        const float a0 = ring[(t + 2)  % 49][tid];
        const float a1 = ring[(t + 3)  % 49][tid];
        const float a2 = ring[(t + 6)  % 49][tid];
        const float a3 = ring[(t + 12) % 49][tid];
        const float a4 = nv;                  // t + 24, just stored
        const float d0 = (t >= 1)  ? ring[(t - 1)  % 49][tid] : 0.0f;
        const float d1 = (t >= 3)  ? ring[(t - 3)  % 49][tid] : 0.0f;
        const float d2 = (t >= 6)  ? ring[(t - 6)  % 49][tid] : 0.0f;
        const float d3 = (t >= 12) ? ring[(t - 12) % 49][tid] : 0.0f;
        const float d4 = (t >= 24) ? ring[(t - 24) % 49][tid] : 0.0f;
        s0 += a0 - d0; s1 += a1 - d1; s2 += a2 - d2; s3 += a3 - d3; s4 += a4 - d4;
    }
}

// ---------------------------------------------------------------------------
// Kernel 5: fused channel-mix GEMM (WMMA f32 16x16x4) + epilogue
//   out[b,o,t,v] = (sum_{k<64} fw[o,k]*cat[k] + fb[o] + x[b,o,t,v])
//                  * (1 + cv[b,t,v]) + impact[b,t]
//   block = (b, t, 64-wide v tile); 8 waves each do a 16(o) x 16(v) tile
// ---------------------------------------------------------------------------
__global__ void fused_kernel(const float* __restrict__ x,
                             const float* __restrict__ comb,
                             const float* __restrict__ cv,
                             const float* __restrict__ imp,
                             const float* __restrict__ fw,
                             const float* __restrict__ fb,
                             float* __restrict__ out) {
    __shared__ float sIn[64][64];   // k=0..31 -> x channels, k=32..63 -> combined
    __shared__ float sW[32][64];    // fusion_w, row-major (o, k)
    __shared__ float sCV[64];
    __shared__ float sFB[32];

    const int tid  = threadIdx.x;
    const int vblk = blockIdx.x % (V / 64);
    const int t    = (blockIdx.x / (V / 64)) % T;
    const int b    = blockIdx.x / ((V / 64) * T);
    const int v0   = vblk * 64;

    for (int i = tid; i < 32 * 64; i += 256) sW[i >> 6][i & 63] = fw[i];
    for (int i = tid; i < 32 * 64; i += 256) {
        const int c = i >> 6, p = i & 63;
        const size_t idx = ((size_t)(b * C + c) * T + t) * V + v0 + p;
        sIn[c][p]      = x[idx];
        sIn[32 + c][p] = comb[idx];
    }
    if (tid < 64) sCV[tid] = cv[((size_t)b * T + t) * V + v0 + tid];
    if (tid < 32) sFB[tid] = fb[tid];
    const float impv = imp[b * T + t];
    __syncthreads();

    const int wave  = tid >> 5;           // 0..7
    const int lane  = tid & 31;
    const int obase = (wave >> 2) * 16;   // 0 or 16
    const int pbase = (wave & 3) * 16;    // 0,16,32,48
    const int m     = lane & 15;
    const int hi    = lane >> 4;

    v8f acc = {};
    #pragma unroll
    for (int kc = 0; kc < 64; kc += 4) {
        const int ka = kc + 2 * hi;       // {kc, kc+2} per half-wave
        v2f a, bm;
        // A (16x4 f32): VGPR0 = K=ka, VGPR1 = K=ka+1 for this half-wave
        a.x = sW[obase + m][ka];
        a.y = sW[obase + m][ka + 1];
        // B (4x16 f32): VGPR0 = row K=ka (lanes hold N), VGPR1 = row K=ka+1
        bm.x = sIn[ka][pbase + m];
        bm.y = sIn[ka + 1][pbase + m];
        acc = __builtin_amdgcn_wmma_f32_16x16x4_f32(
            /*neg_a=*/false, a, /*neg_b=*/false, bm,
            /*c_mod=*/(short)0, acc, /*reuse_a=*/false, /*reuse_b=*/false);
    }

    // D layout: VGPR r -> M = r (lanes 0-15) or r+8 (lanes 16-31), N = lane%16
    const int p = pbase + m;
    const float cvm = 1.0f + sCV[p];
    #pragma unroll
    for (int r = 0; r < 8; ++r) {
        const int o = obase + r + 8 * hi;
        const float val = acc[r] + sFB[o] + sIn[o][p];   // + bias + residual x
        out[((size_t)(b * C + o) * T + t) * V + v0 + p] = val * cvm + impv;
    }
}

// ---------------------------------------------------------------------------
extern "C" void kernel_launch(void* const* d_in, const int* in_sizes, int n_in,
                              void* d_out, int out_size, void* d_ws, size_t ws_size,
                              hipStream_t stream) {
    const float* flow   = (const float*)d_in[0];
    const float* events = (const float*)d_in[1];
    const float* x      = (const float*)d_in[2];
    const float* ef     = (const float*)d_in[3];
    const float* w1     = (const float*)d_in[4];
    const float* b1     = (const float*)d_in[5];
    const float* w2     = (const float*)d_in[6];
    const float* b2     = (const float*)d_in[7];
    const float* fw     = (const float*)d_in[8];
    const float* fb     = (const float*)d_in[9];
    float* out = (float*)d_out;

    float* ws   = (float*)d_ws;
    float* comb = ws;                                   // B*C*T*V
    float* cvb  = comb + (size_t)B * C * T * V;         // B*T*V
    float* impb = cvb  + (size_t)B * T * V;             // B*T
    float* ewb  = impb + (size_t)B * T;                 // B*5

    ew_kernel<<<1, 32, 0, stream>>>(ef, w1, b1, w2, b2, ewb);
    impact_kernel<<<1, 32, 0, stream>>>(events, impb);
    cv_kernel<<<(B * V) / 256, 256, 0, stream>>>(flow, cvb);
    comb_kernel<<<(B * C * V) / 256, 256, 0, stream>>>(x, ewb, comb);
    fused_kernel<<<B * T * (V / 64), 256, 0, stream>>>(x, comb, cvb, impb, fw, fb, out);
}